// GraphBasedSkipConnection_74113955660029
// MI455X (gfx1250) — compile-verified
//
#include <hip/hip_runtime.h>

// MI455X (gfx1250) — wave32, WMMA f32 16x16x4 path (full f32 precision, matches
// the f32 reference). ~21.5 GFLOP of GEMM, matrix-pipe bound; edges computed
// on the fly to avoid materializing the 67MB edge tensor.
// v2: 2x2 register blocking per wave (32x32 output/wave) to double WMMA density
// per LDS fragment read — f32 WMMA (K=4) is operand-bandwidth hungry.

typedef __attribute__((ext_vector_type(2))) float v2f;
typedef __attribute__((ext_vector_type(8))) float v8f;

#define EPSBN 1e-3f
#define MTOT  8192   // B*H*W
#define CCH   512
#define C2CH  256

enum { MODE_RAW = 0, MODE_UV = 1, MODE_EDGE = 2, MODE_FINAL = 3 };

struct GemmArgs {
  const float* x;     // primary A source [8192,512] (k < 512)
  const float* a2;    // secondary A source (agg [*,512] or uf [*,256]) (k >= 512)
  const float* Bw;    // weights, row-major [K, ldB]
  const float* Sbuf;  // shared x@W_lo part, read in edge epilogue
  const float* bias;  // per-col bias (may be null)
  const float* g; const float* b; const float* m; const float* v; // per-col BN
  const float* w_er;  // [4] edge-reduce weights (edge mode)
  float* Y;
  int K, ldB, ldA2, mode;
};

__device__ __forceinline__ int neighbor_pixel(int gm, int nid) {
  int bb = gm >> 12;
  int h  = (gm >> 6) & 63;
  int w  = gm & 63;
  if      (nid == 0) h = (h + 63) & 63;  // roll +1 axis H -> x[h-1]
  else if (nid == 1) h = (h + 1)  & 63;  // roll -1 axis H -> x[h+1]
  else if (nid == 2) w = (w + 63) & 63;  // roll +1 axis W -> x[w-1]
  else               w = (w + 1)  & 63;  // roll -1 axis W -> x[w+1]
  return (bb << 12) | (h << 6) | w;
}

// Epilogue for one 16-row strip (two 16x16 tiles side by side).
__device__ __forceinline__ void epilogue_pair(
    const GemmArgs& p, int gMbase, int g16, int gN0, int gN1, int nid,
    const v8f& acc0, const v8f& acc1) {
  if (p.mode == MODE_RAW) {
    #pragma unroll
    for (int i = 0; i < 8; ++i) {
      const int gM = gMbase + i + 8 * g16;
      p.Y[(size_t)gM * C2CH + gN0] = acc0[i];
      p.Y[(size_t)gM * C2CH + gN1] = acc1[i];
    }
  } else if (p.mode == MODE_EDGE) {
    const float wern = p.w_er[nid];
    const float sc0 = p.g[gN0] * rsqrtf(p.v[gN0] + EPSBN);
    const float sh0 = (p.bias[gN0] - p.m[gN0]) * sc0 + p.b[gN0];
    const float sc1 = p.g[gN1] * rsqrtf(p.v[gN1] + EPSBN);
    const float sh1 = (p.bias[gN1] - p.m[gN1]) * sc1 + p.b[gN1];
    #pragma unroll
    for (int i = 0; i < 8; ++i) {
      const int gM = gMbase + i + 8 * g16;
      const size_t i0 = (size_t)gM * C2CH + gN0;
      const size_t i1 = (size_t)gM * C2CH + gN1;
      const float u0 = fmaxf(0.0f, (acc0[i] + p.Sbuf[i0]) * sc0 + sh0);
      const float u1 = fmaxf(0.0f, (acc1[i] + p.Sbuf[i1]) * sc1 + sh1);
      unsafeAtomicAdd(p.Y + i0, u0 * wern);
      unsafeAtomicAdd(p.Y + i1, u1 * wern);
    }
  } else {  // MODE_UV / MODE_FINAL : Y = relu(bn(acc + bias))
    const int ldY = (p.mode == MODE_UV) ? C2CH : CCH;
    const float bias0 = p.bias ? p.bias[gN0] : 0.0f;
    const float bias1 = p.bias ? p.bias[gN1] : 0.0f;
    const float sc0 = p.g[gN0] * rsqrtf(p.v[gN0] + EPSBN);
    const float sh0 = (bias0 - p.m[gN0]) * sc0 + p.b[gN0];
    const float sc1 = p.g[gN1] * rsqrtf(p.v[gN1] + EPSBN);
    const float sh1 = (bias1 - p.m[gN1]) * sc1 + p.b[gN1];
    #pragma unroll
    for (int i = 0; i < 8; ++i) {
      const int gM = gMbase + i + 8 * g16;
      p.Y[(size_t)gM * ldY + gN0] = fmaxf(0.0f, acc0[i] * sc0 + sh0);
      p.Y[(size_t)gM * ldY + gN1] = fmaxf(0.0f, acc1[i] * sc1 + sh1);
    }
  }
}

__global__ __launch_bounds__(256) void gemm_wmma_f32(GemmArgs p) {
  __shared__ float As[128][36];  // 128 rows x 32 K (padded: stride 36)
  __shared__ float Bs[32][72];   // 32 K x 64 N  (padded: stride 72)

  const int t    = threadIdx.x;
  const int lane = t & 31;
  const int wave = t >> 5;
  const int g16  = lane >> 4;    // lane group 0/1
  const int r    = lane & 15;
  const int wm   = wave & 3;     // 32-row strip within 128
  const int wn   = wave >> 2;    // 32-col half within 64
  const int mBase = blockIdx.y * 128;
  const int nBase = blockIdx.x * 64;
  const int nid   = blockIdx.z;

  v8f acc00 = {};  // (wm*32 +  0..15) x (wn*32 +  0..15)
  v8f acc01 = {};  // (wm*32 +  0..15) x (wn*32 + 16..31)
  v8f acc10 = {};  // (wm*32 + 16..31) x (wn*32 +  0..15)
  v8f acc11 = {};  // (wm*32 + 16..31) x (wn*32 + 16..31)

  const int arow = t >> 3;           // 0..31 (4 passes -> 128 rows)
  const int acol = (t & 7) << 2;     // 0..28
  const int brow = t >> 4;           // 0..15 (2 passes -> 32 rows)
  const int bcol = (t & 15) << 2;    // 0..60

  for (int kb = 0; kb < p.K; kb += 32) {
    // ---- stage A tile (128x32) ----
    #pragma unroll
    for (int rr = 0; rr < 4; ++rr) {
      const int row = arow + rr * 32;
      const int gm  = mBase + row;
      float4 val;
      if (p.mode == MODE_EDGE) {
        const float4 xa = *(const float4*)(p.x + (size_t)gm * CCH + kb + acol);
        const int gpn   = neighbor_pixel(gm, nid);
        const float4 xb = *(const float4*)(p.x + (size_t)gpn * CCH + kb + acol);
        val = make_float4(xa.x * xb.x, xa.y * xb.y, xa.z * xb.z, xa.w * xb.w);
      } else if (kb < CCH) {
        val = *(const float4*)(p.x + (size_t)gm * CCH + kb + acol);
      } else {
        val = *(const float4*)(p.a2 + (size_t)gm * p.ldA2 + (kb - CCH) + acol);
      }
      *(float4*)&As[row][acol] = val;
    }
    // ---- stage B tile (32x64) ----
    #pragma unroll
    for (int rr = 0; rr < 2; ++rr) {
      const int row = brow + rr * 16;
      const float4 wv = *(const float4*)(p.Bw + (size_t)(kb + row) * p.ldB + nBase + bcol);
      *(float4*)&Bs[row][bcol] = wv;
    }
    __syncthreads();

    // ---- 8 WMMA k-steps of K=4 each; 4 WMMAs per step (2x2 reuse) ----
    #pragma unroll
    for (int kk = 0; kk < 32; kk += 4) {
      const int k0 = kk + 2 * g16;
      v2f a0, a1, b0, b1;
      a0.x = As[wm * 32 + r][k0];
      a0.y = As[wm * 32 + r][k0 + 1];
      a1.x = As[wm * 32 + 16 + r][k0];
      a1.y = As[wm * 32 + 16 + r][k0 + 1];
      b0.x = Bs[k0][wn * 32 + r];
      b0.y = Bs[k0 + 1][wn * 32 + r];
      b1.x = Bs[k0][wn * 32 + 16 + r];
      b1.y = Bs[k0 + 1][wn * 32 + 16 + r];
      acc00 = __builtin_amdgcn_wmma_f32_16x16x4_f32(false, a0, false, b0, (short)0, acc00, false, false);
      acc01 = __builtin_amdgcn_wmma_f32_16x16x4_f32(false, a0, false, b1, (short)0, acc01, false, false);
      acc10 = __builtin_amdgcn_wmma_f32_16x16x4_f32(false, a1, false, b0, (short)0, acc10, false, false);
      acc11 = __builtin_amdgcn_wmma_f32_16x16x4_f32(false, a1, false, b1, (short)0, acc11, false, false);
    }
    __syncthreads();
  }

  // ---- epilogue: two 16-row strips, each with two 16-col tiles ----
  const int gN0 = nBase + wn * 32 + r;
  const int gN1 = gN0 + 16;
  epilogue_pair(p, mBase + wm * 32,      g16, gN0, gN1, nid, acc00, acc01);
  epilogue_pair(p, mBase + wm * 32 + 16, g16, gN0, gN1, nid, acc10, acc11);
}

// agg[p,c] = relu(bn1(sum_n w_ea[n]*x[p_n,c]*x[p,c] + b_ea))
__global__ __launch_bounds__(128) void agg_kernel(
    const float* __restrict__ x, const float* __restrict__ w_ea,
    const float* __restrict__ b_ea, const float* __restrict__ g1,
    const float* __restrict__ b1, const float* __restrict__ m1,
    const float* __restrict__ v1, float* __restrict__ agg) {
  const int pix = blockIdx.x;
  const int c   = threadIdx.x << 2;
  const int bb = pix >> 12, h = (pix >> 6) & 63, w = pix & 63;
  const int pu = (bb << 12) | (((h + 63) & 63) << 6) | w;
  const int pd = (bb << 12) | (((h + 1) & 63) << 6) | w;
  const int pl = (bb << 12) | (h << 6) | ((w + 63) & 63);
  const int pr = (bb << 12) | (h << 6) | ((w + 1) & 63);
  const float w0 = w_ea[0], w1 = w_ea[1], w2 = w_ea[2], w3 = w_ea[3];
  const float sc = g1[0] * rsqrtf(v1[0] + EPSBN);
  const float sh = (b_ea[0] - m1[0]) * sc + b1[0];
  const float4 xv = *(const float4*)(x + (size_t)pix * CCH + c);
  const float4 xu = *(const float4*)(x + (size_t)pu  * CCH + c);
  const float4 xd = *(const float4*)(x + (size_t)pd  * CCH + c);
  const float4 xl = *(const float4*)(x + (size_t)pl  * CCH + c);
  const float4 xr = *(const float4*)(x + (size_t)pr  * CCH + c);
  float4 o;
  o.x = fmaxf(0.0f, xv.x * (w0 * xu.x + w1 * xd.x + w2 * xl.x + w3 * xr.x) * sc + sh);
  o.y = fmaxf(0.0f, xv.y * (w0 * xu.y + w1 * xd.y + w2 * xl.y + w3 * xr.y) * sc + sh);
  o.z = fmaxf(0.0f, xv.z * (w0 * xu.z + w1 * xd.z + w2 * xl.z + w3 * xr.z) * sc + sh);
  o.w = fmaxf(0.0f, xv.w * (w0 * xu.w + w1 * xd.w + w2 * xl.w + w3 * xr.w) * sc + sh);
  *(float4*)(agg + (size_t)pix * CCH + c) = o;
}

__global__ __launch_bounds__(256) void zero_kernel(float4* p) {
  p[blockIdx.x * 256 + threadIdx.x] = make_float4(0.f, 0.f, 0.f, 0.f);
}

// uf = uv * relu(bn4(acc + b_er))
__global__ __launch_bounds__(256) void combine_kernel(
    const float* __restrict__ uv, const float* __restrict__ accb,
    const float* __restrict__ b_er, const float* __restrict__ g4,
    const float* __restrict__ b4, const float* __restrict__ m4,
    const float* __restrict__ v4, float* __restrict__ uf) {
  const int i = blockIdx.x * 256 + threadIdx.x;
  const float sc = g4[0] * rsqrtf(v4[0] + EPSBN);
  const float sh = (b_er[0] - m4[0]) * sc + b4[0];
  const float4 a = ((const float4*)accb)[i];
  const float4 u = ((const float4*)uv)[i];
  float4 o;
  o.x = u.x * fmaxf(0.0f, a.x * sc + sh);
  o.y = u.y * fmaxf(0.0f, a.y * sc + sh);
  o.z = u.z * fmaxf(0.0f, a.z * sc + sh);
  o.w = u.w * fmaxf(0.0f, a.w * sc + sh);
  ((float4*)uf)[i] = o;
}

extern "C" void kernel_launch(void* const* d_in, const int* in_sizes, int n_in,
                              void* d_out, int out_size, void* d_ws, size_t ws_size,
                              hipStream_t stream) {
  const float* x    = (const float*)d_in[0];
  const float* w_ea = (const float*)d_in[1];
  const float* b_ea = (const float*)d_in[2];
  const float* g1 = (const float*)d_in[3];
  const float* b1 = (const float*)d_in[4];
  const float* m1 = (const float*)d_in[5];
  const float* v1 = (const float*)d_in[6];
  const float* w_vu = (const float*)d_in[7];
  const float* b_vu = (const float*)d_in[8];
  const float* g2 = (const float*)d_in[9];
  const float* b2 = (const float*)d_in[10];
  const float* m2 = (const float*)d_in[11];
  const float* v2 = (const float*)d_in[12];
  const float* w_eu = (const float*)d_in[13];
  const float* b_eu = (const float*)d_in[14];
  const float* g3 = (const float*)d_in[15];
  const float* b3 = (const float*)d_in[16];
  const float* m3 = (const float*)d_in[17];
  const float* v3 = (const float*)d_in[18];
  const float* w_er = (const float*)d_in[19];
  const float* b_er = (const float*)d_in[20];
  const float* g4 = (const float*)d_in[21];
  const float* b4 = (const float*)d_in[22];
  const float* m4 = (const float*)d_in[23];
  const float* v4 = (const float*)d_in[24];
  const float* w_fc = (const float*)d_in[25];
  const float* g5 = (const float*)d_in[26];
  const float* b5 = (const float*)d_in[27];
  const float* m5 = (const float*)d_in[28];
  const float* v5 = (const float*)d_in[29];

  float* ws   = (float*)d_ws;
  float* agg  = ws;                              // [8192,512]  (dead after uv GEMM)
  float* uf   = ws;                              // [8192,256]  aliases agg
  float* uv   = ws + (size_t)MTOT * CCH;         // [8192,256]
  float* S    = uv + (size_t)MTOT * C2CH;        // [8192,256]
  float* accb = S  + (size_t)MTOT * C2CH;        // [8192,256]

  // 1) zero the edge-reduce accumulator
  zero_kernel<<<(MTOT * C2CH) / 4 / 256, 256, 0, stream>>>((float4*)accb);

  // 2) edge aggregation (elementwise, bandwidth-bound)
  agg_kernel<<<MTOT, 128, 0, stream>>>(x, w_ea, b_ea, g1, b1, m1, v1, agg);

  // 3) uv = relu(bn2([x|agg] @ w_vu + b_vu))   M=8192 K=1024 N=256
  {
    GemmArgs a = {};
    a.x = x; a.a2 = agg; a.Bw = w_vu; a.bias = b_vu;
    a.g = g2; a.b = b2; a.m = m2; a.v = v2;
    a.Y = uv; a.K = 1024; a.ldB = C2CH; a.ldA2 = CCH; a.mode = MODE_UV;
    gemm_wmma_f32<<<dim3(C2CH / 64, MTOT / 128, 1), 256, 0, stream>>>(a);
  }
  // 4) S = x @ w_eu[0:512]  (raw)              M=8192 K=512 N=256
  {
    GemmArgs a = {};
    a.x = x; a.Bw = w_eu; a.Y = S; a.K = CCH; a.ldB = C2CH; a.mode = MODE_RAW;
    gemm_wmma_f32<<<dim3(C2CH / 64, MTOT / 128, 1), 256, 0, stream>>>(a);
  }
  // 5) acc += relu(bn3(S + edge_n @ w_eu[512:1024] + b_eu)) * w_er[n], n=0..3
  {
    GemmArgs a = {};
    a.x = x; a.Bw = w_eu + (size_t)CCH * C2CH; a.Sbuf = S; a.bias = b_eu;
    a.g = g3; a.b = b3; a.m = m3; a.v = v3; a.w_er = w_er;
    a.Y = accb; a.K = CCH; a.ldB = C2CH; a.mode = MODE_EDGE;
    gemm_wmma_f32<<<dim3(C2CH / 64, MTOT / 128, 4), 256, 0, stream>>>(a);
  }
  // 6) uf = uv * relu(bn4(acc + b_er))
  combine_kernel<<<(MTOT * C2CH) / 4 / 256, 256, 0, stream>>>(
      uv, accb, b_er, g4, b4, m4, v4, uf);

  // 7) out = relu(bn5([x|uf] @ w_fc))          M=8192 K=768 N=512
  {
    GemmArgs a = {};
    a.x = x; a.a2 = uf; a.Bw = w_fc;
    a.g = g5; a.b = b5; a.m = m5; a.v = v5;
    a.Y = (float*)d_out; a.K = CCH + C2CH; a.ldB = CCH; a.ldA2 = C2CH;
    a.mode = MODE_FINAL; a.bias = nullptr;
    gemm_wmma_f32<<<dim3(CCH / 64, MTOT / 128, 1), 256, 0, stream>>>(a);
  }
}